// EdgeLoss_31748398252544
// MI455X (gfx1250) — compile-verified
//
#include <hip/hip_runtime.h>
#include <math.h>

// ---- problem geometry (fixed by reference) ----
#define IMG_W 512
#define IMG_H 512
#define NCH   3

// ---- tile geometry ----
#define TW 64
#define TH 32
#define HALO_W (TW + 2)   // 66
#define HALO_H (TH + 2)   // 34
#define THREADS 256

typedef __attribute__((ext_vector_type(2))) float v2f;
typedef __attribute__((ext_vector_type(8))) float v8f;

__device__ __forceinline__ float gray_at(const float* __restrict__ img, int b, int gy, int gx) {
    if (gy < 0 || gy >= IMG_H || gx < 0 || gx >= IMG_W) return 0.0f;   // zero padding
    const size_t hw = (size_t)IMG_H * IMG_W;
    const float* p = img + (size_t)b * NCH * hw + (size_t)gy * IMG_W + gx;
    float r  = __builtin_nontemporal_load(p);
    float g  = __builtin_nontemporal_load(p + hw);
    float bl = __builtin_nontemporal_load(p + 2 * hw);
    return fmaf(0.299f, r, fmaf(0.587f, g, 0.114f * bl));
}

__device__ __forceinline__ float sobel_mag(const float t[HALO_H][HALO_W], int hy, int hx) {
    float c00 = t[hy-1][hx-1], c01 = t[hy-1][hx], c02 = t[hy-1][hx+1];
    float c10 = t[hy  ][hx-1],                    c12 = t[hy  ][hx+1];
    float c20 = t[hy+1][hx-1], c21 = t[hy+1][hx], c22 = t[hy+1][hx+1];
    // cross-correlation with SOBEL_X / SOBEL_Y, zero-padded
    float ex = (c02 + 2.0f * c12 + c22) - (c00 + 2.0f * c10 + c20);
    float ey = (c20 + 2.0f * c21 + c22) - (c00 + 2.0f * c01 + c02);
    return sqrtf(ex * ex + ey * ey);
}

// Stage 1: per-tile |edge(pred)-edge(target)| partial sums (bandwidth-bound path).
__global__ __launch_bounds__(THREADS)
void edge_tile_kernel(const float* __restrict__ pred,
                      const float* __restrict__ targ,
                      float* __restrict__ partial) {
    __shared__ float gp[HALO_H][HALO_W];
    __shared__ float gt[HALO_H][HALO_W];
    __shared__ float red[THREADS];

    const int tid = threadIdx.x;
    const int x0 = blockIdx.x * TW;
    const int y0 = blockIdx.y * TH;
    const int b  = blockIdx.z;

    // Stage grayscale halo tiles for both images (coalesced NT global loads).
    for (int i = tid; i < HALO_H * HALO_W; i += THREADS) {
        const int hy = i / HALO_W, hx = i % HALO_W;
        const int gy = y0 + hy - 1, gx = x0 + hx - 1;
        gp[hy][hx] = gray_at(pred, b, gy, gx);
        gt[hy][hx] = gray_at(targ, b, gy, gx);
    }
    __syncthreads();

    float acc = 0.0f;
    for (int i = tid; i < TH * TW; i += THREADS) {
        const int py = i / TW, px = i % TW;       // lanes -> consecutive px: conflict-free LDS
        const int hy = py + 1, hx = px + 1;
        const float ep = sobel_mag(gp, hy, hx);
        const float et = sobel_mag(gt, hy, hx);
        acc += fabsf(ep - et);
    }

    // Deterministic in-block tree reduction.
    red[tid] = acc;
    __syncthreads();
    for (int s = THREADS / 2; s > 0; s >>= 1) {
        if (tid < s) red[tid] += red[tid + s];
        __syncthreads();
    }
    if (tid == 0) {
        const int blk = (blockIdx.z * gridDim.y + blockIdx.y) * gridDim.x + blockIdx.x;
        partial[blk] = red[0];
    }
}

// Stage 2: single-wave WMMA reduction of the block partials.
// A = ones(16x4); each V_WMMA_F32_16X16X4_F32 folds 64 partials (as B) into C.
// sum(D) == 16 * sum(B) regardless of B's lane layout, so the final lane-sum / 16
// is exact and deterministic.
//
// n is guaranteed a multiple of 64 by the stage-1 grid (128 blocks per batch
// image), so the loads are unguarded: tight global_load_b64 + v_wmma loop,
// EXEC all-ones throughout (single full wave, no divergence).
__global__ __launch_bounds__(32)
void wmma_reduce_kernel(const float* __restrict__ partial, int n, float inv_count,
                        float* __restrict__ out) {
    const int lane = threadIdx.x;   // 0..31, one full wave

    v2f a; a.x = 1.0f; a.y = 1.0f;  // all-ones A matrix (16x4 f32 = 2 VGPRs/lane)
    v8f c = {};

    const v2f* pv = (const v2f*)partial;   // 8-byte aligned pairs
    const int npairs = n >> 1;             // pairs; 32 pairs consumed per WMMA
    for (int p = lane; p < npairs; p += 32) {
        const v2f bv = pv[p];
        c = __builtin_amdgcn_wmma_f32_16x16x4_f32(
                /*neg_a=*/false, a, /*neg_b=*/false, bv,
                /*c_mod=*/(short)0, c, /*reuse_a=*/false, /*reuse_b=*/false);
    }

    float s = c[0] + c[1] + c[2] + c[3] + c[4] + c[5] + c[6] + c[7];

    __shared__ float buf[32];
    buf[lane] = s;
    __syncthreads();
    if (lane == 0) {
        float t = 0.0f;
        for (int i = 0; i < 32; ++i) t += buf[i];   // fixed order: deterministic
        out[0] = t * (1.0f / 16.0f) * inv_count;
    }
}

extern "C" void kernel_launch(void* const* d_in, const int* in_sizes, int n_in,
                              void* d_out, int out_size, void* d_ws, size_t ws_size,
                              hipStream_t stream) {
    const float* pred = (const float*)d_in[0];
    const float* targ = (const float*)d_in[1];
    float* out = (float*)d_out;
    float* partial = (float*)d_ws;

    const int B = in_sizes[0] / (NCH * IMG_H * IMG_W);   // 32 for the reference

    dim3 grid(IMG_W / TW, IMG_H / TH, B);                // 8 x 16 x B = 128*B blocks
    edge_tile_kernel<<<grid, THREADS, 0, stream>>>(pred, targ, partial);

    const int nblk = grid.x * grid.y * grid.z;           // multiple of 64 by construction
    const float inv_count = 1.0f / ((float)B * IMG_H * IMG_W);
    wmma_reduce_kernel<<<1, 32, 0, stream>>>(partial, nblk, inv_count, out);
}